// Sinkhorn_50551765074308
// MI455X (gfx1250) — compile-verified
//
#include <hip/hip_runtime.h>
#include <math.h>

#define NDIM 2048
#define KDIM 32
#define ITERS 50

typedef __attribute__((ext_vector_type(2))) float v2f;
typedef __attribute__((ext_vector_type(4))) float f4;
typedef __attribute__((ext_vector_type(8))) float v8f;

// ---------------- helpers ----------------

__device__ __forceinline__ void lse_combine(float& m, float& s, float m2, float s2) {
    float nm = fmaxf(m, m2);
    if (nm == -INFINITY) return;           // both empty
    s = s * __expf(m - nm) + s2 * __expf(m2 - nm);
    m = nm;
}

// ---------------- kernels ----------------

// zero log_alpha (stored) and lrs, vectorized
__global__ __launch_bounds__(256) void sk_init(f4* __restrict__ la4, f4* __restrict__ lrs4) {
    size_t idx = (size_t)blockIdx.x * 256 + threadIdx.x;
    f4 z = {0.f, 0.f, 0.f, 0.f};
    if (idx < (size_t)NDIM * NDIM / 4) la4[idx] = z;
    if (idx < NDIM / 4) lrs4[idx] = z;
}

// xsq[i] = sum_k x[i,k]^2 ; ysq likewise
__global__ __launch_bounds__(256) void sk_sq(const float* __restrict__ x, const float* __restrict__ y,
                                             float* __restrict__ xsq, float* __restrict__ ysq) {
    int idx = blockIdx.x * 256 + threadIdx.x;
    if (idx < NDIM) {
        const f4* r = (const f4*)(x + (size_t)idx * KDIM);
        float a = 0.f;
#pragma unroll
        for (int k = 0; k < KDIM / 4; ++k) {
            f4 v = r[k];
            a += v.x * v.x + v.y * v.y + v.z * v.z + v.w * v.w;
        }
        xsq[idx] = a;
    } else if (idx < 2 * NDIM) {
        int j = idx - NDIM;
        const f4* r = (const f4*)(y + (size_t)j * KDIM);
        float a = 0.f;
#pragma unroll
        for (int k = 0; k < KDIM / 4; ++k) {
            f4 v = r[k];
            a += v.x * v.x + v.y * v.y + v.z * v.z + v.w * v.w;
        }
        ysq[j] = a;
    }
}

// cost[i,j] = xsq[i] + ysq[j] - 2 * (x @ y^T)[i,j]  via f32 WMMA 16x16x4, K=32 (8 steps)
// block = 256 threads = 8 waves; each wave computes one 16x16 tile.
__global__ __launch_bounds__(256) void sk_cost_wmma(const float* __restrict__ x, const float* __restrict__ y,
                                                    const float* __restrict__ xsq, const float* __restrict__ ysq,
                                                    float* __restrict__ cost) {
    const int lane = threadIdx.x & 31;
    const int wave = threadIdx.x >> 5;
    const int m    = lane & 15;
    const int half = lane >> 4;

    const int colTile = blockIdx.x * 8 + wave;   // 128 column tiles / 8 waves per block
    const int rowTile = blockIdx.y;              // 128 row tiles
    const int row0 = rowTile * 16;
    const int col0 = colTile * 16;

    const float* xr = x + (size_t)(row0 + m) * KDIM;  // A: lane m holds row (row0+m)
    const float* yr = y + (size_t)(col0 + m) * KDIM;  // B(col-major): lane m holds y-row (col0+m)

    v8f acc = {};
#pragma unroll
    for (int t = 0; t < 8; ++t) {
        const int kb = t * 4 + half * 2;
        v2f a = { xr[kb], xr[kb + 1] };
        v2f b = { yr[kb], yr[kb + 1] };
        acc = __builtin_amdgcn_wmma_f32_16x16x4_f32(false, a, false, b, (short)0, acc, false, false);
    }

    const int n = m;                                  // D layout: VGPR r, lane -> M = r + 8*half, N = lane&15
    const float ysn = ysq[col0 + n];
#pragma unroll
    for (int r = 0; r < 8; ++r) {
        const int M = r + 8 * half;
        const float c = xsq[row0 + M] + ysn - 2.0f * acc[r];
        cost[(size_t)(row0 + M) * NDIM + (col0 + n)] = c;
    }
}

// row_lse[i] = logsumexp_j( stored[i,j] ) - lrs[i]     (effective la = stored - lrs[i])
__global__ __launch_bounds__(256) void sk_row_lse(const float* __restrict__ la, const float* __restrict__ lrs,
                                                  float* __restrict__ rlse) {
    __shared__ float sm[256];
    __shared__ float ss[256];
    const int i = blockIdx.x;
    const int t = threadIdx.x;
    const f4* row = (const f4*)(la + (size_t)i * NDIM);   // 512 float4 per row

    float m = -INFINITY, s = 0.f;
#pragma unroll
    for (int p = 0; p < 2; ++p) {
        f4 v = row[t + p * 256];
#pragma unroll
        for (int k = 0; k < 4; ++k) lse_combine(m, s, v[k], 1.0f);
    }

    sm[t] = m; ss[t] = s;
    __syncthreads();
    for (int w = 128; w > 0; w >>= 1) {
        if (t < w) {
            float mm = sm[t], sv = ss[t];
            lse_combine(mm, sv, sm[t + w], ss[t + w]);
            sm[t] = mm; ss[t] = sv;
        }
        __syncthreads();
    }
    if (t == 0) rlse[i] = sm[0] + __logf(ss[0]) - lrs[i];
}

// stage 1 of column logsumexp: each thread owns 4 adjacent columns over a 128-row chunk
__global__ __launch_bounds__(256) void sk_col_lse_partial(const float* __restrict__ la, const float* __restrict__ lrs,
                                                          float* __restrict__ pm, float* __restrict__ ps) {
    const int j4 = blockIdx.x * 256 + threadIdx.x;  // float4 column group (0..511), grid.x = 2
    const int chunk = blockIdx.y;                   // 16 chunks of 128 rows
    const int i0 = chunk * 128;
    const f4* lap = (const f4*)la;                  // row stride = NDIM/4 = 512

    float m[4], s[4];
#pragma unroll
    for (int k = 0; k < 4; ++k) { m[k] = -INFINITY; s[k] = 0.f; }

    for (int i = i0; i < i0 + 128; ++i) {
        const f4 v = lap[(size_t)i * (NDIM / 4) + j4];
        const float l = lrs[i];
#pragma unroll
        for (int k = 0; k < 4; ++k) lse_combine(m[k], s[k], v[k] - l, 1.0f);
    }
    f4 mo = { m[0], m[1], m[2], m[3] };
    f4 so = { s[0], s[1], s[2], s[3] };
    ((f4*)pm)[(size_t)chunk * (NDIM / 4) + j4] = mo;
    ((f4*)ps)[(size_t)chunk * (NDIM / 4) + j4] = so;
}

// stage 2: combine the 16 partials per column (4 columns per thread)
__global__ __launch_bounds__(256) void sk_col_lse_combine(const float* __restrict__ pm, const float* __restrict__ ps,
                                                          float* __restrict__ clse) {
    const int j4 = blockIdx.x * 256 + threadIdx.x;  // grid.x = 2
    float m[4], s[4];
#pragma unroll
    for (int k = 0; k < 4; ++k) { m[k] = -INFINITY; s[k] = 0.f; }
#pragma unroll
    for (int c = 0; c < 16; ++c) {
        const f4 vm = ((const f4*)pm)[(size_t)c * (NDIM / 4) + j4];
        const f4 vs = ((const f4*)ps)[(size_t)c * (NDIM / 4) + j4];
#pragma unroll
        for (int k = 0; k < 4; ++k) lse_combine(m[k], s[k], vm[k], vs[k]);
    }
    f4 outv = { m[0] + __logf(s[0]), m[1] + __logf(s[1]), m[2] + __logf(s[2]), m[3] + __logf(s[3]) };
    ((f4*)clse)[j4] = outv;
}

// fused: stored' = (stored - lrs[i]) - 0.5*(rlse[i] + clse[j]);
//        rowsum  = sum_j exp(-cost - stored');  lrs'[i] = log(rowsum)
__global__ __launch_bounds__(256) void sk_update(const float* __restrict__ cost, float* __restrict__ la,
                                                 float* __restrict__ lrs, const float* __restrict__ rlse,
                                                 const float* __restrict__ clse) {
    __shared__ float red[256];
    const int i = blockIdx.x;
    const int t = threadIdx.x;
    const float l0 = lrs[i];
    const float rl = rlse[i];
    const size_t base = (size_t)i * NDIM;
    const f4* cst = (const f4*)(cost + base);
    f4* lap = (f4*)(la + base);
    const f4* cl4 = (const f4*)clse;

    float acc = 0.f;
#pragma unroll
    for (int p = 0; p < 2; ++p) {
        const int jv = t + p * 256;
        f4 v = lap[jv];
        const f4 c = cst[jv];
        const f4 cl = cl4[jv];
        f4 outv;
#pragma unroll
        for (int k = 0; k < 4; ++k) {
            const float nv = v[k] - l0 - 0.5f * (rl + cl[k]);
            outv[k] = nv;
            acc += __expf(-c[k] - nv);
        }
        lap[jv] = outv;
    }
    red[t] = acc;
    __syncthreads();
    for (int w = 128; w > 0; w >>= 1) {
        if (t < w) red[t] += red[t + w];
        __syncthreads();
    }
    if (t == 0) lrs[i] = __logf(red[0]);
}

// per-row partial of final loss: sum_j exp(stored - lrs[i]) * cost
__global__ __launch_bounds__(256) void sk_loss_partial(const float* __restrict__ cost, const float* __restrict__ la,
                                                       const float* __restrict__ lrs, float* __restrict__ part) {
    __shared__ float red[256];
    const int i = blockIdx.x;
    const int t = threadIdx.x;
    const float l0 = lrs[i];
    const size_t base = (size_t)i * NDIM;
    const f4* cst = (const f4*)(cost + base);
    const f4* lap = (const f4*)(la + base);

    float acc = 0.f;
#pragma unroll
    for (int p = 0; p < 2; ++p) {
        const int jv = t + p * 256;
        const f4 v = lap[jv];
        const f4 c = cst[jv];
#pragma unroll
        for (int k = 0; k < 4; ++k) acc += __expf(v[k] - l0) * c[k];
    }
    red[t] = acc;
    __syncthreads();
    for (int w = 128; w > 0; w >>= 1) {
        if (t < w) red[t] += red[t + w];
        __syncthreads();
    }
    if (t == 0) part[i] = red[0];
}

// deterministic single-block reduction of the 2048 row partials
__global__ __launch_bounds__(256) void sk_loss_final(const float* __restrict__ part, float* __restrict__ out) {
    __shared__ float red[256];
    const int t = threadIdx.x;
    float acc = 0.f;
#pragma unroll
    for (int j = t; j < NDIM; j += 256) acc += part[j];
    red[t] = acc;
    __syncthreads();
    for (int w = 128; w > 0; w >>= 1) {
        if (t < w) red[t] += red[t + w];
        __syncthreads();
    }
    if (t == 0) out[0] = red[0];
}

// ---------------- launch ----------------

extern "C" void kernel_launch(void* const* d_in, const int* in_sizes, int n_in,
                              void* d_out, int out_size, void* d_ws, size_t ws_size,
                              hipStream_t stream) {
    (void)in_sizes; (void)n_in; (void)out_size; (void)ws_size;
    const float* x = (const float*)d_in[0];
    const float* y = (const float*)d_in[1];
    float* out = (float*)d_out;

    const size_t NN = (size_t)NDIM * NDIM;
    float* W    = (float*)d_ws;
    float* cost = W;                 // NN
    float* la   = W + NN;            // NN (stored log_alpha, lazily offset by lrs)
    float* xsq  = W + 2 * NN;        // NDIM
    float* ysq  = xsq + NDIM;        // NDIM
    float* lrs  = ysq + NDIM;        // NDIM
    float* rlse = lrs + NDIM;        // NDIM
    float* clse = rlse + NDIM;       // NDIM
    float* pm   = clse + NDIM;       // 16*NDIM
    float* ps   = pm + 16 * NDIM;    // 16*NDIM
    float* part = ps + 16 * NDIM;    // NDIM

    // init log_alpha = 0, lrs = 0
    sk_init<<<(unsigned)(NN / 4 / 256), 256, 0, stream>>>((f4*)la, (f4*)lrs);
    // squared norms
    sk_sq<<<(2 * NDIM + 255) / 256, 256, 0, stream>>>(x, y, xsq, ysq);
    // cost matrix via WMMA GEMM
    {
        dim3 grid(NDIM / (16 * 8), NDIM / 16);   // (16, 128)
        sk_cost_wmma<<<grid, 256, 0, stream>>>(x, y, xsq, ysq, cost);
    }

    // Sinkhorn iterations
    for (int it = 0; it < ITERS; ++it) {
        sk_row_lse<<<NDIM, 256, 0, stream>>>(la, lrs, rlse);
        {
            dim3 grid(NDIM / 1024, 16);          // (2, 16)
            sk_col_lse_partial<<<grid, 256, 0, stream>>>(la, lrs, pm, ps);
        }
        sk_col_lse_combine<<<NDIM / 1024, 256, 0, stream>>>(pm, ps, clse);
        sk_update<<<NDIM, 256, 0, stream>>>(cost, la, lrs, rlse, clse);
    }

    // loss = sum P * cost
    sk_loss_partial<<<NDIM, 256, 0, stream>>>(cost, la, lrs, part);
    sk_loss_final<<<1, 256, 0, stream>>>(part, out);
}